// HRGATConv_5068061409834
// MI455X (gfx1250) — compile-verified
//
#include <hip/hip_runtime.h>

#define HID 128
#define N_P 30000
#define N_A 30000
#define E_P 480000
#define E_A 480000
#define E_X 480000
#define NEG_SLOPE 0.2f
#define LN_EPS 1e-5f

typedef __attribute__((ext_vector_type(2))) float v2f;
typedef __attribute__((ext_vector_type(8))) float v8f;

#if defined(__has_builtin)
#if __has_builtin(__builtin_amdgcn_wmma_f32_16x16x4_f32)
#define USE_WMMA_F32 1
#endif
#endif

// ---------- device helpers ----------
__device__ __forceinline__ float wave_sum(float v) {
#pragma unroll
  for (int o = 16; o > 0; o >>= 1) v += __shfl_xor(v, o, 32);
  return v;
}
__device__ __forceinline__ int f2ord(float f) {
  int i = __float_as_int(f);
  return (i >= 0) ? i : (i ^ 0x7FFFFFFF);
}
__device__ __forceinline__ float ord2f(int i) {
  return __int_as_float((i >= 0) ? i : (i ^ 0x7FFFFFFF));
}
__device__ __forceinline__ float geluf(float v) {
  return 0.5f * v * (1.f + erff(v * 0.70710678118654752440f));
}

// ---------- GEMM: out[m][n] = sum_k A[m][k] * W[n][k]   (M%16==0, K=N=128) ----------
// WMMA f32 16x16x4. A-frag (ISA 7.12.2, 32-bit A 16x4): lane l -> M=l%16, K=2*(l/16)+r.
// B-frag (4x16, K x N): lane l -> N=l%16, K=2*(l/16)+r. Since out = A @ W^T, the K x N
// oriented B tile is exactly row-major W: B[k][n] = W[n][k] -> contiguous v2f per lane.
// C/D: vgpr r, lane l -> M = r + 8*(l/16), N = l%16.
__global__ __launch_bounds__(256) void gemm_xWT(const float* __restrict__ A,
                                                const float* __restrict__ W,
                                                float* __restrict__ out, int M) {
  const int wave = threadIdx.x >> 5;  // 0..7 -> 16-col slice of N=128
  const int lane = threadIdx.x & 31;
  const int half = lane >> 4;
  const int l16 = lane & 15;
  const int m0 = blockIdx.x << 4;
  const int n0 = wave << 4;
#ifdef USE_WMMA_F32
  const float* arow = A + (size_t)(m0 + l16) * HID + 2 * half;
  const float* brow = W + (size_t)(n0 + l16) * HID + 2 * half;
  v8f acc = {0.f, 0.f, 0.f, 0.f, 0.f, 0.f, 0.f, 0.f};
#pragma unroll
  for (int kk = 0; kk < HID; kk += 4) {
    v2f av = *reinterpret_cast<const v2f*>(arow + kk);
    v2f bv = *reinterpret_cast<const v2f*>(brow + kk);
    acc = __builtin_amdgcn_wmma_f32_16x16x4_f32(false, av, false, bv, (short)0, acc,
                                                false, false);
  }
  float* orow = out + (size_t)(m0 + half * 8) * HID + n0 + l16;
#pragma unroll
  for (int r = 0; r < 8; ++r) orow[(size_t)r * HID] = acc[r];
#else
  // scalar fallback (compile-safety only): 2048 outputs per block, 8 per thread
  for (int i = 0; i < 8; ++i) {
    int idx = threadIdx.x + 256 * i;          // 0..2047
    int m = m0 + (idx >> 7);
    int n = idx & 127;
    float s = 0.f;
    for (int k = 0; k < HID; ++k) s += A[(size_t)m * HID + k] * W[(size_t)n * HID + k];
    out[(size_t)m * HID + n] = s;
  }
  (void)wave; (void)lane; (void)half; (void)l16; (void)n0;
#endif
}

// ---------- out[t*N+n] = dot(feat[n,:], vecs[t*vstride+voff : +128]) ----------
__global__ __launch_bounds__(256) void rowdot_multi(const float* __restrict__ feat,
                                                    const float* __restrict__ vecs,
                                                    int vstride, int voff,
                                                    float* __restrict__ out, int N, int T) {
  int wave = blockIdx.x * (blockDim.x >> 5) + (threadIdx.x >> 5);
  int lane = threadIdx.x & 31;
  if (wave >= N) return;
  float4 xv = *reinterpret_cast<const float4*>(feat + (size_t)wave * HID + lane * 4);
  for (int t = 0; t < T; ++t) {
    const float* v = vecs + (size_t)t * vstride + voff + lane * 4;
    float p = xv.x * v[0] + xv.y * v[1] + xv.z * v[2] + xv.w * v[3];
    p = wave_sum(p);
    if (lane == 0) out[(size_t)t * N + wave] = p;
  }
}

__global__ void fill_i32(int* p, int v, int n) {
  int i = blockIdx.x * blockDim.x + threadIdx.x;
  if (i < n) p[i] = v;
}

// ---------- edge pass 1: raw score + segment max (ordered-int atomicMax) ----------
__global__ __launch_bounds__(256) void edge_score_max(
    const int* __restrict__ src, const int* __restrict__ dst, const int* __restrict__ et,
    const float* __restrict__ sDst, const float* __restrict__ sSrc,
    float* __restrict__ esc, int* __restrict__ mx, int E, int Ndst, int Nsrc) {
  int e = blockIdx.x * blockDim.x + threadIdx.x;
  if (e >= E) return;
  int t = et[e];
  float s = sDst[(size_t)t * Ndst + dst[e]] + sSrc[(size_t)t * Nsrc + src[e]];
  esc[e] = s;
  atomicMax(&mx[(size_t)t * Ndst + dst[e]], f2ord(s));
}

// ---------- edge pass 2: exp(score - max) + segment sum ----------
__global__ __launch_bounds__(256) void edge_exp_sum(const int* __restrict__ dst,
                                                    const int* __restrict__ et,
                                                    const int* __restrict__ mx,
                                                    float* __restrict__ esc,
                                                    float* __restrict__ den, int E, int Ndst) {
  int e = blockIdx.x * blockDim.x + threadIdx.x;
  if (e >= E) return;
  int t = et[e];
  int d = dst[e];
  float ev = expf(esc[e] - ord2f(mx[(size_t)t * Ndst + d]));
  esc[e] = ev;
  atomicAdd(&den[(size_t)t * Ndst + d], ev);
}

// ---------- edge pass 3: agg[t][dst] += feat[src] * alpha ; 32 threads / edge ----------
__global__ __launch_bounds__(256) void edge_aggregate(
    const int* __restrict__ src, const int* __restrict__ dst, const int* __restrict__ et,
    const float* __restrict__ esc, const float* __restrict__ den,
    const float* __restrict__ feat, float* __restrict__ agg, int E, int Ndst) {
  int gid = blockIdx.x * blockDim.x + threadIdx.x;
  int e = gid >> 5;
  if (e >= E) return;
  int c = (gid & 31) * 4;
  int t = et[e];
  int d = dst[e];
  int s = src[e];
  float dn = den[(size_t)t * Ndst + d];
  float a = esc[e] / (dn > 0.f ? dn : 1.f);
  float4 xv = *reinterpret_cast<const float4*>(feat + (size_t)s * HID + c);
  float* ap = agg + ((size_t)t * Ndst + d) * HID + c;
  atomicAdd(ap + 0, xv.x * a);
  atomicAdd(ap + 1, xv.y * a);
  atomicAdd(ap + 2, xv.z * a);
  atomicAdd(ap + 3, xv.w * a);
}

// ---------- LayerNorm + leaky (neg=1.0 -> plain LN) ; one wave per row ----------
__global__ __launch_bounds__(256) void ln_leaky(const float* __restrict__ in,
                                                const float* __restrict__ g,
                                                const float* __restrict__ b,
                                                float* __restrict__ out, int N, float neg) {
  int wave = blockIdx.x * (blockDim.x >> 5) + (threadIdx.x >> 5);
  int lane = threadIdx.x & 31;
  if (wave >= N) return;
  float4 xv = *reinterpret_cast<const float4*>(in + (size_t)wave * HID + lane * 4);
  float mu = wave_sum(xv.x + xv.y + xv.z + xv.w) * (1.f / HID);
  float dx = xv.x - mu, dy = xv.y - mu, dz = xv.z - mu, dw = xv.w - mu;
  float var = wave_sum(dx * dx + dy * dy + dz * dz + dw * dw) * (1.f / HID);
  float r = rsqrtf(var + LN_EPS);
  float4 gv = *reinterpret_cast<const float4*>(g + lane * 4);
  float4 bv = *reinterpret_cast<const float4*>(b + lane * 4);
  float4 y;
  y.x = dx * r * gv.x + bv.x;
  y.y = dy * r * gv.y + bv.y;
  y.z = dz * r * gv.z + bv.z;
  y.w = dw * r * gv.w + bv.w;
  y.x = y.x > 0.f ? y.x : neg * y.x;
  y.y = y.y > 0.f ? y.y : neg * y.y;
  y.z = y.z > 0.f ? y.z : neg * y.z;
  y.w = y.w > 0.f ? y.w : neg * y.w;
  *reinterpret_cast<float4*>(out + (size_t)wave * HID + lane * 4) = y;
}

__global__ __launch_bounds__(256) void leaky_ew(const float* __restrict__ in,
                                                float* __restrict__ out, int n4, float neg) {
  int i = blockIdx.x * blockDim.x + threadIdx.x;
  if (i >= n4) return;
  float4 v = reinterpret_cast<const float4*>(in)[i];
  v.x = v.x > 0.f ? v.x : neg * v.x;
  v.y = v.y > 0.f ? v.y : neg * v.y;
  v.z = v.z > 0.f ? v.z : neg * v.z;
  v.w = v.w > 0.f ? v.w : neg * v.w;
  reinterpret_cast<float4*>(out)[i] = v;
}

// ---------- beta mix + gelu/skip + LayerNorm ; one wave per node ----------
__global__ __launch_bounds__(256) void combine_kernel(
    const float* __restrict__ xskip, const float* __restrict__ outs,
    const float* __restrict__ bsc, const float* __restrict__ ob_p,
    const float* __restrict__ bw_p, const float* __restrict__ skip_p,
    const float* __restrict__ ng, const float* __restrict__ nb, float* __restrict__ hout,
    int N, int T, int use_ob) {
  int wave = blockIdx.x * (blockDim.x >> 5) + (threadIdx.x >> 5);
  int lane = threadIdx.x & 31;
  if (wave >= N) return;
  float bs[4];
  float m = -3.4e38f;
  for (int t = 0; t < T; ++t) {
    float v = bsc[(size_t)t * N + wave];
    bs[t] = v;
    m = fmaxf(m, v);
  }
  float z = 0.f;
  for (int t = 0; t < T; ++t) { bs[t] = expf(bs[t] - m); z += bs[t]; }
  float wc[4];
  if (use_ob) {
    float obv[4];
    float om = -3.4e38f;
    for (int t = 0; t < T; ++t) { obv[t] = ob_p[t]; om = fmaxf(om, obv[t]); }
    float oz = 0.f;
    for (int t = 0; t < T; ++t) { obv[t] = expf(obv[t] - om); oz += obv[t]; }
    float bw = 1.f / (1.f + expf(-bw_p[0]));
    for (int t = 0; t < T; ++t) wc[t] = (1.f - bw) * (bs[t] / z) + bw * (obv[t] / oz);
  } else {
    for (int t = 0; t < T; ++t) wc[t] = bs[t] / z;
  }
  float fw = 1.f / (1.f + expf(-skip_p[0]));
  float4 res = make_float4(0.f, 0.f, 0.f, 0.f);
  for (int t = 0; t < T; ++t) {
    float4 ov = *reinterpret_cast<const float4*>(outs + ((size_t)t * N + wave) * HID + lane * 4);
    res.x += ov.x * wc[t];
    res.y += ov.y * wc[t];
    res.z += ov.z * wc[t];
    res.w += ov.w * wc[t];
  }
  float4 xv = *reinterpret_cast<const float4*>(xskip + (size_t)wave * HID + lane * 4);
  float4 y;
  y.x = geluf(res.x) * fw + xv.x * (1.f - fw);
  y.y = geluf(res.y) * fw + xv.y * (1.f - fw);
  y.z = geluf(res.z) * fw + xv.z * (1.f - fw);
  y.w = geluf(res.w) * fw + xv.w * (1.f - fw);
  float mu = wave_sum(y.x + y.y + y.z + y.w) * (1.f / HID);
  float dx = y.x - mu, dy = y.y - mu, dz = y.z - mu, dw = y.w - mu;
  float var = wave_sum(dx * dx + dy * dy + dz * dz + dw * dw) * (1.f / HID);
  float r = rsqrtf(var + LN_EPS);
  float4 gv = *reinterpret_cast<const float4*>(ng + lane * 4);
  float4 bv = *reinterpret_cast<const float4*>(nb + lane * 4);
  float4 o;
  o.x = dx * r * gv.x + bv.x;
  o.y = dy * r * gv.y + bv.y;
  o.z = dz * r * gv.z + bv.z;
  o.w = dw * r * gv.w + bv.w;
  *reinterpret_cast<float4*>(hout + (size_t)wave * HID + lane * 4) = o;
}

// ================= host side =================
struct Ws {
  float *hp0, *hp1, *ha0, *ha1, *x, *outs, *tmp;
  float *sA, *sB, *den, *bsc, *esc;
  int* mx;
};

static inline int cdiv(int a, int b) { return (a + b - 1) / b; }
#define ORD_NEG_INF ((int)0x807FFFFFu)

// DGAT layer: xin [N,128] -> hout [N,128]
static void run_dgat(const float* xin, float* hout, const int* eidx, const int* etype, int E,
                     void* const* P, int pb, int N, int T, const Ws& w, hipStream_t s) {
  const float* bw_p   = (const float*)P[pb + 0];
  const float* ln_b   = (const float*)P[pb + 1];
  const float* ln_g   = (const float*)P[pb + 2];
  const float* norm_b = (const float*)P[pb + 3];
  const float* norm_g = (const float*)P[pb + 4];
  const float* ob     = (const float*)P[pb + 5];
  const float* rel_bt = (const float*)P[pb + 6];
  const float* rel_wi = (const float*)P[pb + 7];
  const float* skip   = (const float*)P[pb + 8];
  const float* w_bt   = (const float*)P[pb + 9];
  const float* w_wi   = (const float*)P[pb + 10];
  const int* src = eidx;
  const int* dst = eidx + E;
  size_t NH = (size_t)N * HID;

  gemm_xWT<<<N / 16, 256, 0, s>>>(xin, w_wi, w.x, N);
  rowdot_multi<<<cdiv(N, 8), 256, 0, s>>>(w.x, rel_wi, 2 * HID, 0, w.sA, N, T);    // dst proj
  rowdot_multi<<<cdiv(N, 8), 256, 0, s>>>(w.x, rel_wi, 2 * HID, HID, w.sB, N, T);  // src proj
  hipMemsetAsync(w.outs, 0, (size_t)T * NH * sizeof(float), s);
  hipMemsetAsync(w.den, 0, (size_t)T * N * sizeof(float), s);
  fill_i32<<<cdiv(T * N, 256), 256, 0, s>>>(w.mx, ORD_NEG_INF, T * N);
  edge_score_max<<<cdiv(E, 256), 256, 0, s>>>(src, dst, etype, w.sA, w.sB, w.esc, w.mx, E, N, N);
  edge_exp_sum<<<cdiv(E, 256), 256, 0, s>>>(dst, etype, w.mx, w.esc, w.den, E, N);
  edge_aggregate<<<cdiv(E * 32, 256), 256, 0, s>>>(src, dst, etype, w.esc, w.den, w.x, w.outs, E, N);
  for (int t = 0; t < T; ++t) {
    ln_leaky<<<cdiv(N, 8), 256, 0, s>>>(w.outs + (size_t)t * NH, ln_g + (size_t)t * HID,
                                        ln_b + (size_t)t * HID, w.tmp, N, NEG_SLOPE);
    gemm_xWT<<<N / 16, 256, 0, s>>>(w.tmp, w_bt, w.outs + (size_t)t * NH, N);
    // softmax over types is shift-invariant -> the x@rb_i term cancels; only rs@rb_o matters
    rowdot_multi<<<cdiv(N, 8), 256, 0, s>>>(w.outs + (size_t)t * NH, rel_bt, 0, HID,
                                            w.bsc + (size_t)t * N, N, 1);
  }
  combine_kernel<<<cdiv(N, 8), 256, 0, s>>>(w.x, w.outs, w.bsc, ob, bw_p, skip, norm_g, norm_b,
                                            hout, N, T, 1);
}

// HetGAT layer: src features afeat [Nsrc,128], dst features pfeat [Ndst,128] -> pout
static void run_hetgat(const float* afeat, const float* pfeat, float* pout, const int* src,
                       const int* dst, const int* etype, int E, void* const* P, int pb,
                       int Ndst, int Nsrc, const Ws& w, hipStream_t s) {
  const float* norm_b = (const float*)P[pb + 0];
  const float* norm_g = (const float*)P[pb + 1];
  const float* rel_bt = (const float*)P[pb + 2];
  const float* rel_wi = (const float*)P[pb + 3];
  const float* skip   = (const float*)P[pb + 4];
  const float* w_bt   = (const float*)P[pb + 5];
  const int T = 2;
  size_t NH = (size_t)Ndst * HID;

  rowdot_multi<<<cdiv(Ndst, 8), 256, 0, s>>>(pfeat, rel_wi, 2 * HID, 0, w.sA, Ndst, T);
  rowdot_multi<<<cdiv(Nsrc, 8), 256, 0, s>>>(afeat, rel_wi, 2 * HID, HID, w.sB, Nsrc, T);
  hipMemsetAsync(w.outs, 0, (size_t)T * NH * sizeof(float), s);
  hipMemsetAsync(w.den, 0, (size_t)T * Ndst * sizeof(float), s);
  fill_i32<<<cdiv(T * Ndst, 256), 256, 0, s>>>(w.mx, ORD_NEG_INF, T * Ndst);
  edge_score_max<<<cdiv(E, 256), 256, 0, s>>>(src, dst, etype, w.sA, w.sB, w.esc, w.mx, E, Ndst, Nsrc);
  edge_exp_sum<<<cdiv(E, 256), 256, 0, s>>>(dst, etype, w.mx, w.esc, w.den, E, Ndst);
  edge_aggregate<<<cdiv(E * 32, 256), 256, 0, s>>>(src, dst, etype, w.esc, w.den, afeat, w.outs, E, Ndst);
  for (int t = 0; t < T; ++t) {
    leaky_ew<<<cdiv((int)(NH / 4), 256), 256, 0, s>>>(w.outs + (size_t)t * NH, w.tmp,
                                                      (int)(NH / 4), NEG_SLOPE);
    gemm_xWT<<<Ndst / 16, 256, 0, s>>>(w.tmp, w_bt, w.outs + (size_t)t * NH, Ndst);
    rowdot_multi<<<cdiv(Ndst, 8), 256, 0, s>>>(w.outs + (size_t)t * NH, rel_bt, 0, HID,
                                               w.bsc + (size_t)t * Ndst, Ndst, 1);
  }
  combine_kernel<<<cdiv(Ndst, 8), 256, 0, s>>>(pfeat, w.outs, w.bsc, nullptr, nullptr, skip,
                                               norm_g, norm_b, pout, Ndst, T, 0);
}

extern "C" void kernel_launch(void* const* d_in, const int* in_sizes, int n_in, void* d_out,
                              int out_size, void* d_ws, size_t ws_size, hipStream_t stream) {
  (void)in_sizes; (void)n_in; (void)out_size; (void)ws_size;
  const float* h_p = (const float*)d_in[0];
  const float* h_a = (const float*)d_in[1];
  const int* pei = (const int*)d_in[2];
  const int* pet = (const int*)d_in[3];
  const int* aei = (const int*)d_in[4];
  const int* aet = (const int*)d_in[5];
  const int* xei = (const int*)d_in[6];
  const int* xet = (const int*)d_in[7];
  // pytree leaf order: dicts flatten with sorted keys.
  // dgat dict (11 leaves): beta_weight, ln_b, ln_g, norm_b, norm_g, overall_beta,
  //                        rel_bt, rel_wi, skip, w_bt, w_wi       -> bases 8,19,30,41
  // hetgat dict (6 leaves): norm_b, norm_g, rel_bt, rel_wi, skip, w_bt -> bases 52,58,64,70
  const float* fg = (const float*)d_in[76];
  const float* fb = (const float*)d_in[77];

  // workspace layout (floats); needs ~158 MB
  float* f = (float*)d_ws;
  const size_t NH = (size_t)N_P * HID;  // N_P == N_A
  Ws w;
  w.hp0 = f;            w.hp1 = f + NH;
  w.ha0 = f + 2 * NH;   w.ha1 = f + 3 * NH;
  w.x = f + 4 * NH;
  w.outs = f + 5 * NH;  // 4*NH
  w.tmp = f + 9 * NH;
  float* ext = f + 10 * NH;
  w.sA = ext;                 // 4*30000
  w.sB = ext + 120000;
  w.den = ext + 240000;
  w.bsc = ext + 360000;
  w.mx = (int*)(ext + 480000);
  w.esc = ext + 600000;       // 480000

  // ---- 2 interleaved DGAT rounds ----
  run_dgat(h_p,   w.hp0, pei, pet, E_P, d_in, 8 + 11 * 0, N_P, 4, w, stream);
  run_dgat(h_a,   w.ha0, aei, aet, E_A, d_in, 8 + 11 * 1, N_A, 4, w, stream);
  run_dgat(w.hp0, w.hp0, pei, pet, E_P, d_in, 8 + 11 * 2, N_P, 4, w, stream);
  run_dgat(w.ha0, w.ha0, aei, aet, E_A, d_in, 8 + 11 * 3, N_A, 4, w, stream);

  // ---- 2 HetGAT rounds (p_new from old (h_a,h_p); a_new from old (h_p,h_a)) ----
  // round 0: read hp0/ha0, write hp1/ha1
  run_hetgat(w.ha0, w.hp0, w.hp1, xei, xei + E_X, xet, E_X, d_in, 52 + 6 * 0, N_P, N_A, w, stream);
  run_hetgat(w.hp0, w.ha0, w.ha1, xei + E_X, xei, xet, E_X, d_in, 52 + 6 * 1, N_A, N_P, w, stream);
  // round 1: read hp1/ha1, write hp0/ha0
  run_hetgat(w.ha1, w.hp1, w.hp0, xei, xei + E_X, xet, E_X, d_in, 52 + 6 * 2, N_P, N_A, w, stream);
  run_hetgat(w.hp1, w.ha1, w.ha0, xei + E_X, xei, xet, E_X, d_in, 52 + 6 * 3, N_A, N_P, w, stream);

  // ---- final LayerNorms: return (LN(h_a), LN(h_p)) concatenated ----
  float* out = (float*)d_out;
  ln_leaky<<<cdiv(N_A, 8), 256, 0, stream>>>(w.ha0, fg, fb, out, N_A, 1.0f);
  ln_leaky<<<cdiv(N_P, 8), 256, 0, stream>>>(w.hp0, fg, fb, out + (size_t)N_A * HID, N_P, 1.0f);
}